// GradientConsistencyLoss_45535243272458
// MI455X (gfx1250) — compile-verified
//
#include <hip/hip_runtime.h>
#include <hip/hip_bf16.h>
#include <stdint.h>

typedef __attribute__((ext_vector_type(2))) float v2f;
typedef __attribute__((ext_vector_type(8))) float v8f;

#define TILE_H   16
#define ROWS     (TILE_H + 2)     // halo rows per strip
#define IMG      512
#define DW       516              // padded width of difference tile (cols -1..512)
#define NTHREADS 256

// Pass 1: per-strip fused (channel-mean diff -> sobel x/y -> sum |.|) partials.
__global__ __launch_bounds__(NTHREADS) void sobel_partial_kernel(
    const float* __restrict__ pred, const float* __restrict__ gt,
    float* __restrict__ partials)
{
    __shared__ __align__(16) float raw_p[ROWS * IMG];   // halo rows x 512
    __shared__ __align__(16) float raw_g[ROWS * IMG];
    __shared__ float dt[ROWS * DW];                     // 3x channel-mean difference
    __shared__ float wred[NTHREADS / 32];

    const int tid    = threadIdx.x;
    const int bid    = blockIdx.x;
    const int strips = IMG / TILE_H;                    // 32
    const int n      = bid / strips;
    const int strip  = bid - n * strips;
    const int r0     = strip * TILE_H;

    // zero difference tile (also provides zero padding for halo rows/cols)
    for (int i = tid; i < ROWS * DW; i += NTHREADS) dt[i] = 0.0f;

    const unsigned lds_p = (unsigned)(uintptr_t)(&raw_p[0]);
    const unsigned lds_g = (unsigned)(uintptr_t)(&raw_g[0]);

    for (int ch = 0; ch < 3; ++ch) {
        const unsigned plane = (unsigned)(n * 3 + ch) * (unsigned)(IMG * IMG);
        // async-copy ROWS halo rows x 512 floats of pred & gt into LDS.
        // 16B per lane; NT hint: 201MB single-use stream > 192MB L2, zero reuse.
        for (int t = 0; t < 2; ++t) {
            const float* base  = t ? gt : pred;
            const unsigned lb  = t ? lds_g : lds_p;
            for (int it = 0; it < (ROWS * 128) / NTHREADS; ++it) {   // 9
                int chunk = tid + it * NTHREADS;      // 0..ROWS*128-1, 16B chunks
                int ri    = chunk >> 7;               // row in tile 0..ROWS-1
                int c16   = chunk & 127;              // 16B chunk within row
                int grow  = r0 - 1 + ri;
                if (grow >= 0 && grow < IMG) {
                    unsigned goff = ((plane + (unsigned)grow * IMG) << 2) + ((unsigned)c16 << 4);
                    unsigned loff = lb + ((unsigned)(ri * IMG) << 2) + ((unsigned)c16 << 4);
                    asm volatile("global_load_async_to_lds_b128 %0, %1, %2 th:TH_LOAD_NT"
                                 :: "v"(loff), "v"(goff), "s"(base)
                                 : "memory");
                }
            }
        }
#if __has_builtin(__builtin_amdgcn_s_wait_asynccnt)
        __builtin_amdgcn_s_wait_asynccnt(0);
#else
        asm volatile("s_wait_asynccnt 0x0" ::: "memory");
#endif
        __syncthreads();
        // accumulate 3x-scaled difference into dt (col shifted by +1 for halo)
        for (int it = 0; it < (ROWS * IMG) / NTHREADS; ++it) {       // 36
            int e    = tid + it * NTHREADS;
            int ri   = e >> 9;
            int c    = e & (IMG - 1);
            int grow = r0 - 1 + ri;
            if (grow >= 0 && grow < IMG)
                dt[ri * DW + c + 1] += raw_p[e] - raw_g[e];
        }
        __syncthreads();
    }

    // sobel X/Y from LDS; accumulate |gx|+|gy|
    float acc = 0.0f;
    for (int i = 0; i < (TILE_H * IMG) / NTHREADS; ++i) {            // 32
        int p = tid + i * NTHREADS;
        int r = p >> 9;
        int c = p & (IMG - 1);
        const float* t = &dt[r * DW + c];
        const float* m = t + DW;
        const float* b = m + DW;
        float t0 = t[0], t1 = t[1], t2 = t[2];
        float m0 = m[0],             m2 = m[2];
        float b0 = b[0], b1 = b[1], b2 = b[2];
        float gx = (t2 - t0) + 2.0f * (m2 - m0) + (b2 - b0);
        float gy = (b0 - t0) + 2.0f * (b1 - t1) + (b2 - t2);
        acc += fabsf(gx) + fabsf(gy);
    }

    // wave reduction via V_WMMA_F32_16X16X4_F32: D = A x ones
    // A[m,0]=acc(lane m), A[m,2]=acc(lane m+16) -> row m sum = acc(m)+acc(m+16)
    // lane L<16 holds D[0..7, L] in its 8 VGPRs; lane L>=16 holds D[8..15, .]
    float wave_total;
#if __has_builtin(__builtin_amdgcn_wmma_f32_16x16x4_f32)
    {
        v2f a;   a.x = acc;  a.y = 0.0f;
        v2f one; one.x = 1.0f; one.y = 1.0f;
        v8f cz = {};
        v8f d = __builtin_amdgcn_wmma_f32_16x16x4_f32(
            false, a, false, one, (short)0, cz, false, false);
        float s = ((d[0] + d[1]) + (d[2] + d[3])) + ((d[4] + d[5]) + (d[6] + d[7]));
        wave_total = s + __shfl_xor(s, 16, 32);   // rows 0-7 sum + rows 8-15 sum
    }
#else
    {
        float s = acc;
        for (int off = 16; off > 0; off >>= 1) s += __shfl_xor(s, off, 32);
        wave_total = s;
    }
#endif
    if ((tid & 31) == 0) wred[tid >> 5] = wave_total;
    __syncthreads();
    if (tid == 0) {
        float s = 0.0f;
        for (int w = 0; w < NTHREADS / 32; ++w) s += wred[w];
        partials[bid] = s;
    }
}

// Pass 2: deterministic fixed-order reduction of block partials -> scalar loss.
__global__ __launch_bounds__(NTHREADS) void reduce_kernel(
    const float* __restrict__ partials, int n, float* __restrict__ out, float scale)
{
    __shared__ float s[NTHREADS];
    float a = 0.0f;
    for (int i = threadIdx.x; i < n; i += NTHREADS) a += partials[i];
    s[threadIdx.x] = a;
    __syncthreads();
    for (int w = NTHREADS / 2; w > 0; w >>= 1) {
        if (threadIdx.x < w) s[threadIdx.x] += s[threadIdx.x + w];
        __syncthreads();
    }
    if (threadIdx.x == 0) out[0] = s[0] * scale;
}

extern "C" void kernel_launch(void* const* d_in, const int* in_sizes, int n_in,
                              void* d_out, int out_size, void* d_ws, size_t ws_size,
                              hipStream_t stream)
{
    const float* pred = (const float*)d_in[0];
    const float* gt   = (const float*)d_in[1];
    const int N       = in_sizes[0] / (3 * IMG * IMG);   // 32
    const int strips  = IMG / TILE_H;                    // 32
    const int nblocks = N * strips;                      // 1024
    float* partials   = (float*)d_ws;                    // nblocks * 4 bytes

    sobel_partial_kernel<<<nblocks, NTHREADS, 0, stream>>>(pred, gt, partials);

    // dt holds 3x the channel-mean difference; loss = sum/(3 * N*H*W)
    const float scale = 1.0f / (3.0f * (float)N * (float)IMG * (float)IMG);
    reduce_kernel<<<1, NTHREADS, 0, stream>>>(partials, nblocks, (float*)d_out, scale);
}